// Pllay2_91276644974928
// MI455X (gfx1250) — compile-verified
//
#include <hip/hip_runtime.h>
#include <hip/hip_bf16.h>

// ---------------------------------------------------------------------------
// Problem constants (match reference)
// ---------------------------------------------------------------------------
#define HH 28
#define WW 28
#define NPTS 784          // H*W
#define BATCH 64
#define TT 32             // landscape resolution
#define KK1 2
#define KK2 3
#define F1W (2*KK1*TT)    // 128
#define F2W (2*KK2*TT)    // 192

typedef float v2f __attribute__((ext_vector_type(2)));
typedef float v8f __attribute__((ext_vector_type(8)));

// Workspace byte offsets
#define OFF_SI     0u
#define OFF_D2S    (OFF_SI   + (size_t)NPTS*NPTS*4)
#define OFF_F1     (OFF_D2S  + (size_t)NPTS*NPTS*4)
#define OFF_F2     (OFF_F1   + (size_t)BATCH*NPTS*4)
#define OFF_BI1    (OFF_F2   + (size_t)BATCH*NPTS*4)
#define OFF_DI1    (OFF_BI1  + (size_t)BATCH*2*NPTS*4)
#define OFF_BI2    (OFF_DI1  + (size_t)BATCH*2*NPTS*4)
#define OFF_DI2    (OFF_BI2  + (size_t)BATCH*2*NPTS*4)
#define OFF_FEAT1  (OFF_DI2  + (size_t)BATCH*2*NPTS*4)
#define OFF_FEAT2  (OFF_FEAT1+ (size_t)BATCH*F1W*4)
#define OFF_X12    (OFF_FEAT2+ (size_t)BATCH*F2W*4)
#define WS_NEEDED  (OFF_X12  + (size_t)BATCH*64*4)

__device__ const int NB_DX[8] = {-1, 1, 0, 0, -1, -1, 1, 1};
__device__ const int NB_DY[8] = { 0, 0,-1, 1, -1,  1,-1, 1};

// ---------------------------------------------------------------------------
// Kernel 1: recompute _SORT_IDX / _D2S. Grid spacing is exactly 1.0, so
// squared distances are small ints; key = d2*1024 + j is a distinct total
// order reproducing numpy's *stable* argsort. Rank-by-count sort per row.
// ---------------------------------------------------------------------------
__global__ void grid_keys_kernel(int* __restrict__ si, float* __restrict__ d2s) {
  __shared__ int key[NPTS];
  const int i  = blockIdx.x;
  const int ix = i / WW, iy = i % WW;
  for (int j = threadIdx.x; j < NPTS; j += blockDim.x) {
    int jx = j / WW, jy = j % WW;
    int dx = ix - jx, dy = iy - jy;
    key[j] = (dx*dx + dy*dy) * 1024 + j;
  }
  __syncthreads();
  for (int j = threadIdx.x; j < NPTS; j += blockDim.x) {
    const int kj = key[j];
    int rank = 0;
    for (int k = 0; k < NPTS; ++k) rank += (key[k] < kj) ? 1 : 0;
    si [i*NPTS + rank] = j;
    d2s[i*NPTS + rank] = (float)(kj >> 10);
  }
}

// ---------------------------------------------------------------------------
// Kernel 2: DTM for both m0 values fused. One workgroup per sample, weights
// in LDS; inner scan early-exits once cumulative mass passes the larger bound
// (weights are strictly positive -> all later eff terms are zero). The sorted
// index / distance rows are streamed sequentially, so prefetch ahead.
// ---------------------------------------------------------------------------
__global__ void dtm_kernel(const float* __restrict__ x,
                           const int* __restrict__ si,
                           const float* __restrict__ d2s,
                           float* __restrict__ f1, float* __restrict__ f2) {
  __shared__ float xl[NPTS];
  __shared__ float partial[256];
  const int m = blockIdx.x;
  const float* xm = x + (size_t)m * NPTS;
  float s = 0.f;
  for (int j = threadIdx.x; j < NPTS; j += blockDim.x) { float v = xm[j]; xl[j] = v; s += v; }
  partial[threadIdx.x] = s;
  __syncthreads();
  for (int off = 128; off > 0; off >>= 1) {
    if ((int)threadIdx.x < off) partial[threadIdx.x] += partial[threadIdx.x + off];
    __syncthreads();
  }
  const float total  = partial[0];
  const float bound1 = 0.05f * total;
  const float bound2 = 0.20f * total;
  for (int i = threadIdx.x; i < NPTS; i += blockDim.x) {
    const int*   sir = si  + (size_t)i * NPTS;
    const float* dr  = d2s + (size_t)i * NPTS;
    __builtin_prefetch(sir, 0, 3);        // global_prefetch_b8: row start
    __builtin_prefetch(dr,  0, 3);
    float cum = 0.f, acc1 = 0.f, acc2 = 0.f;
    for (int j = 0; j < NPTS; ++j) {
      if ((j & 31) == 0) {                // prefetch a cacheline ahead
        __builtin_prefetch(sir + j + 32, 0, 3);
        __builtin_prefetch(dr  + j + 32, 0, 3);
      }
      const float prev = cum;             // cum - w of the reference
      if (prev >= bound2) break;          // both eff streams are 0 forever
      const float w = xl[sir[j]];
      cum = prev + w;
      const float d = dr[j];
      acc1 += fminf(fmaxf(bound1 - prev, 0.f), w) * d;
      acc2 += fminf(fmaxf(bound2 - prev, 0.f), w) * d;
    }
    f1[(size_t)m*NPTS + i] = sqrtf(acc1 / bound1);
    f2[(size_t)m*NPTS + i] = sqrtf(acc2 / bound2);
  }
}

// ---------------------------------------------------------------------------
// Kernel 3: H0 persistence pairs. 256 tasks = {f1,f2} x 64 samples x
// {x/4-conn, -x/8-conn}. Parallel stable rank-sort in LDS, then thread 0
// runs the reference union-find exactly. For filt==1 the (bi,di) outputs are
// swapped, matching `bi[m,1], di[m,1] = (d1, b1)`.
// ---------------------------------------------------------------------------
__global__ void pairs_kernel(const float* __restrict__ f1, const float* __restrict__ f2,
                             int* __restrict__ bi1, int* __restrict__ di1,
                             int* __restrict__ bi2, int* __restrict__ di2) {
  __shared__ float v[NPTS];
  __shared__ int   pos[NPTS];
  __shared__ int   order[NPTS];
  __shared__ int   parent[NPTS];
  __shared__ int   birth[NPTS];
  __shared__ int   added[NPTS];

  const int task  = blockIdx.x;      // 0..255
  const int which = task >> 7;       // 0 -> f1, 1 -> f2
  const int m     = (task >> 1) & 63;
  const int filt  = task & 1;        // 0: x 4-conn, 1: -x 8-conn
  const float sgn = filt ? -1.f : 1.f;
  const float* vals = (which ? f2 : f1) + (size_t)m * NPTS;

  for (int j = threadIdx.x; j < NPTS; j += blockDim.x) {
    v[j] = sgn * vals[j];
    added[j] = 0;
  }
  __syncthreads();
  for (int j = threadIdx.x; j < NPTS; j += blockDim.x) {
    const float vj = v[j];
    int r = 0;
    for (int k = 0; k < NPTS; ++k) {
      const float vk = v[k];
      r += (vk < vj || (vk == vj && k < j)) ? 1 : 0;   // stable sort rank
    }
    pos[j] = r;
    order[r] = j;
    parent[j] = j;
    birth[j] = j;
  }
  __syncthreads();

  const int base = (m*2 + filt) * NPTS;
  int* bo = (which ? bi2 : bi1) + base;
  int* dd = (which ? di2 : di1) + base;
  for (int j = threadIdx.x; j < NPTS; j += blockDim.x) { bo[j] = 0; dd[j] = 0; }
  __syncthreads();

  if (threadIdx.x == 0) {
    const int nn = filt ? 8 : 4;
    int kcnt = 0;
    for (int t = 0; t < NPTS; ++t) {
      const int vtx = order[t];
      added[vtx] = 1;
      const int vi = vtx / WW, vj2 = vtx % WW;
      for (int q = 0; q < nn; ++q) {
        const int ui = vi + NB_DX[q], uj = vj2 + NB_DY[q];
        if (ui < 0 || ui >= HH || uj < 0 || uj >= WW) continue;
        const int u = ui*WW + uj;
        if (!added[u]) continue;
        int ru = u;   while (parent[ru] != ru) ru = parent[ru];
        { int xx = u;   while (parent[xx] != ru) { int nx = parent[xx]; parent[xx] = ru; xx = nx; } }
        int rv = vtx; while (parent[rv] != rv) rv = parent[rv];
        { int xx = vtx; while (parent[xx] != rv) { int nx = parent[xx]; parent[xx] = rv; xx = nx; } }
        if (ru == rv) continue;
        int lb;
        if (pos[birth[ru]] <= pos[birth[rv]]) { lb = birth[rv]; parent[rv] = ru; }
        else                                  { lb = birth[ru]; parent[ru] = rv; }
        if (kcnt < NPTS) {
          if (filt == 0) { bo[kcnt] = lb;  dd[kcnt] = vtx; }
          else           { bo[kcnt] = vtx; dd[kcnt] = lb;  }   // H1 swap
          ++kcnt;
        }
      }
    }
  }
}

// ---------------------------------------------------------------------------
// Kernel 4: landscape features. One thread per (which, m, filt, t): scan all
// 784 pairs keeping the top-3 tent values (K1=2 uses the first two).
// Feature layout: feat[m, (filt*K + k)*T + t]  (matches reference reshape).
// ---------------------------------------------------------------------------
__global__ void landscape_kernel(const float* __restrict__ f1, const float* __restrict__ f2,
                                 const int* __restrict__ bi1, const int* __restrict__ di1,
                                 const int* __restrict__ bi2, const int* __restrict__ di2,
                                 float* __restrict__ feat1, float* __restrict__ feat2) {
  const int tid = blockIdx.x * blockDim.x + threadIdx.x;
  if (tid >= 2*BATCH*2*TT) return;
  const int which = tid / (BATCH*2*TT);
  const int rem   = tid % (BATCH*2*TT);
  const int m     = rem / (2*TT);
  const int filt  = (rem / TT) & 1;
  const int ti    = rem % TT;

  const float* vals = (which ? f2 : f1) + (size_t)m * NPTS;
  const int*   bo   = (which ? bi2 : bi1) + (m*2 + filt) * NPTS;
  const int*   dd   = (which ? di2 : di1) + (m*2 + filt) * NPTS;
  const float start = which ? 1.f : 0.f;          // (0,7) or (1,8): span is 7
  const float t = start + 7.f * (float)ti / (float)(TT - 1);

  float top0 = 0.f, top1 = 0.f, top2 = 0.f;       // tents are clamped >= 0
  for (int p = 0; p < NPTS; ++p) {
    const float b = vals[bo[p]];
    const float d = vals[dd[p]];
    float tent = fmaxf(fminf(t - b, d - t), 0.f);
    if      (tent > top0) { top2 = top1; top1 = top0; top0 = tent; }
    else if (tent > top1) { top2 = top1; top1 = tent; }
    else if (tent > top2) { top2 = tent; }
  }
  if (which == 0) {
    float* ft = feat1 + (size_t)m * F1W;
    ft[(filt*KK1 + 0)*TT + ti] = top0;
    ft[(filt*KK1 + 1)*TT + ti] = top1;
  } else {
    float* ft = feat2 + (size_t)m * F2W;
    ft[(filt*KK2 + 0)*TT + ti] = top0;
    ft[(filt*KK2 + 1)*TT + ti] = top1;
    ft[(filt*KK2 + 2)*TT + ti] = top2;
  }
}

// ---------------------------------------------------------------------------
// WMMA tile helper: one wave computes a 16x16 tile of relu(A @ W^T + b).
// K is a compile-time constant so the loop gets *scalar* control flow — EXEC
// stays all-ones across the whole v_wmma_f32_16x16x4_f32 stream (ISA §7.12
// requirement), no per-iteration exec masking.
// Fragment layouts per ISA 7.12.2:
//   A 16x4 : lanes 0-15 -> {K=k0,k0+1}, lanes 16-31 -> {K=k0+2,k0+3}
//   B 4x16 : VGPR r holds row K=k0+r (lanes 0-15) / K=k0+r+2 (lanes 16-31)
//   C/D    : VGPR r = (M=r, N=lane) for lanes 0-15, (M=r+8, N=lane-16) above.
// ---------------------------------------------------------------------------
template <int K>
__device__ __forceinline__ void wmma_tile_relu(const float* __restrict__ A,
                                               const float* __restrict__ Wg,
                                               const float* __restrict__ bg,
                                               float* __restrict__ x12,
                                               int mt, int nt, int lane, int colbase) {
  const int lane16 = lane & 15;
  const int hi     = lane >> 4;
  const int arow   = mt*16 + lane16;
  const int bcol   = nt*16 + lane16;

  v8f acc = {};
#pragma unroll 4
  for (int k0 = 0; k0 < K; k0 += 4) {
    const int ka = k0 + hi*2;
    v2f a = { A [(size_t)arow*K + ka], A [(size_t)arow*K + ka + 1] };
    v2f b = { Wg[(size_t)bcol*K + ka], Wg[(size_t)bcol*K + ka + 1] };   // B = Wg^T
    acc = __builtin_amdgcn_wmma_f32_16x16x4_f32(false, a, false, b,
                                                (short)0, acc, false, false);
  }
  const float bias = bg[bcol];
  const int col = colbase + bcol;
#pragma unroll
  for (int r = 0; r < 8; ++r) {
    const int row = mt*16 + r + hi*8;
    x12[(size_t)row*64 + col] = fmaxf(acc[r] + bias, 0.f);
  }
}

// Kernel 5: both hidden layers. 16 waves: wave = (stage, mtile, ntile).
__global__ void gemm1_wmma_kernel(const float* __restrict__ feat1, const float* __restrict__ feat2,
                                  const float* __restrict__ w_g1, const float* __restrict__ b_g1,
                                  const float* __restrict__ w_g2, const float* __restrict__ b_g2,
                                  float* __restrict__ x12) {
  const int wave  = threadIdx.x >> 5;
  const int lane  = threadIdx.x & 31;
  const int stage = wave >> 3;            // 0: g1(K=128), 1: g2(K=192)
  const int mt    = (wave >> 1) & 3;      // 4 row tiles of 16
  const int nt    = wave & 1;             // 2 col tiles of 16

  if (stage == 0) {
    wmma_tile_relu<F1W>(feat1, w_g1, b_g1, x12, mt, nt, lane, 0);
  } else {
    wmma_tile_relu<F2W>(feat2, w_g2, b_g2, x12, mt, nt, lane, 32);
  }
}

// Kernel 6: output layer (64x64)@(64x10) with N padded to one 16-wide tile.
__global__ void gemm2_wmma_kernel(const float* __restrict__ x12,
                                  const float* __restrict__ w_fc,
                                  const float* __restrict__ b_fc,
                                  float* __restrict__ out) {
  const int mt     = threadIdx.x >> 5;    // 4 waves = 4 row tiles
  const int lane   = threadIdx.x & 31;
  const int lane16 = lane & 15;
  const int hi     = lane >> 4;
  const int arow   = mt*16 + lane16;
  const int n      = lane16;              // valid columns n < 10

  v8f acc = {};
#pragma unroll 4
  for (int k0 = 0; k0 < 64; k0 += 4) {
    const int ka = k0 + hi*2;
    v2f a = { x12[(size_t)arow*64 + ka], x12[(size_t)arow*64 + ka + 1] };
    v2f b = { (n < 10) ? w_fc[(size_t)n*64 + ka]     : 0.f,
              (n < 10) ? w_fc[(size_t)n*64 + ka + 1] : 0.f };
    acc = __builtin_amdgcn_wmma_f32_16x16x4_f32(false, a, false, b,
                                                (short)0, acc, false, false);
  }
  if (n < 10) {
    const float bias = b_fc[n];
#pragma unroll
    for (int r = 0; r < 8; ++r) {
      const int row = mt*16 + r + hi*8;
      out[(size_t)row*10 + n] = acc[r] + bias;
    }
  }
}

// ---------------------------------------------------------------------------
extern "C" void kernel_launch(void* const* d_in, const int* in_sizes, int n_in,
                              void* d_out, int out_size, void* d_ws, size_t ws_size,
                              hipStream_t stream) {
  if (ws_size < WS_NEEDED) return;

  const float* input = (const float*)d_in[0];
  const float* w_g1  = (const float*)d_in[1];
  const float* b_g1  = (const float*)d_in[2];
  const float* w_g2  = (const float*)d_in[3];
  const float* b_g2  = (const float*)d_in[4];
  const float* w_fc  = (const float*)d_in[5];
  const float* b_fc  = (const float*)d_in[6];
  float* out = (float*)d_out;

  char* ws = (char*)d_ws;
  int*   si    = (int*)  (ws + OFF_SI);
  float* d2s   = (float*)(ws + OFF_D2S);
  float* f1    = (float*)(ws + OFF_F1);
  float* f2    = (float*)(ws + OFF_F2);
  int*   bi1   = (int*)  (ws + OFF_BI1);
  int*   di1   = (int*)  (ws + OFF_DI1);
  int*   bi2   = (int*)  (ws + OFF_BI2);
  int*   di2   = (int*)  (ws + OFF_DI2);
  float* feat1 = (float*)(ws + OFF_FEAT1);
  float* feat2 = (float*)(ws + OFF_FEAT2);
  float* x12   = (float*)(ws + OFF_X12);

  grid_keys_kernel<<<NPTS, 256, 0, stream>>>(si, d2s);
  dtm_kernel      <<<BATCH, 256, 0, stream>>>(input, si, d2s, f1, f2);
  pairs_kernel    <<<256, 256, 0, stream>>>(f1, f2, bi1, di1, bi2, di2);
  landscape_kernel<<<(2*BATCH*2*TT + 255)/256, 256, 0, stream>>>(
      f1, f2, bi1, di1, bi2, di2, feat1, feat2);
  gemm1_wmma_kernel<<<1, 512, 0, stream>>>(feat1, feat2, w_g1, b_g1, w_g2, b_g2, x12);
  gemm2_wmma_kernel<<<1, 128, 0, stream>>>(x12, w_fc, b_fc, out);
}